// Dice_Contrastive_Loss_77738908057903
// MI455X (gfx1250) — compile-verified
//
#include <hip/hip_runtime.h>
#include <hip/hip_bf16.h>

typedef __attribute__((ext_vector_type(2))) float v2f;
typedef __attribute__((ext_vector_type(8))) float v8f;

#define TAU 0.1f
#define DICE_SMOOTH 0.1f
#define BATCH 16
#define NPER 262144            // 1*512*512 elements per batch image
#define GRID 512
#define BLOCK 256
#define WAVES (GRID * BLOCK / 32)   // 4096
#define KPW (NPER / WAVES)          // 64 k-values per wave (contiguous segment)
#define ITERS (KPW / 8)             // 8 iterations, 8 k's per iteration

// workspace layout (floats):
// [0..255]   gram  (raw sum f1[i,:].f2[j,:], row-major 16x16)
// [256..271] sum(s1^2) per batch
// [272..287] sum(s2^2) per batch
// [288..303] sum((mask*(s1-s2))^2) per batch
// [304] sum(p*g)  [305] sum(p)  [306] sum(g)
#define WS_FLOATS 307

__device__ __forceinline__ float sigf(float x) {
    // single v_rcp_f32 instead of the precise-division expansion
    return __builtin_amdgcn_rcpf(1.0f + __expf(-x));
}

__global__ void zero_ws_kernel(float* ws) {
    int i = blockIdx.x * blockDim.x + threadIdx.x;
    if (i < WS_FLOATS) ws[i] = 0.0f;
}

__global__ __launch_bounds__(BLOCK)
void fused_reduce_kernel(const float* __restrict__ pred,
                         const float* __restrict__ gt,
                         const float* __restrict__ in1,
                         const float* __restrict__ in2,
                         const float* __restrict__ mask,
                         float* __restrict__ ws) {
    __shared__ float sGram[256];
    __shared__ float sSq1[16];
    __shared__ float sSq2[16];
    __shared__ float sMD[16];
    __shared__ float sDice[3];

    for (int i = threadIdx.x; i < 256; i += blockDim.x) sGram[i] = 0.0f;
    if (threadIdx.x < 16) {
        sSq1[threadIdx.x] = 0.0f;
        sSq2[threadIdx.x] = 0.0f;
        sMD[threadIdx.x]  = 0.0f;
    }
    if (threadIdx.x < 3) sDice[threadIdx.x] = 0.0f;
    __syncthreads();

    const int lane  = threadIdx.x & 31;
    const int row   = lane & 15;        // M index for A, N index for B
    const int khalf = lane >> 4;        // which half of each 8-float chunk
    const int wave  = (blockIdx.x * blockDim.x + threadIdx.x) >> 5;

    // lane's base: row start + wave's contiguous 64-float segment + half offset
    const int base = row * NPER + wave * KPW + khalf * 4;

    // two independent accumulators -> half-depth WMMA dependency chain
    v8f acc0 = {};
    v8f acc1 = {};
    float sq1 = 0.0f, sq2 = 0.0f, md = 0.0f;
    float spg = 0.0f, sp = 0.0f, sg = 0.0f;

    // ---- software-pipelined streaming loop: 5 x b128 per 8 k-values ----
    float4 x1 = *reinterpret_cast<const float4*>(in1  + base);
    float4 x2 = *reinterpret_cast<const float4*>(in2  + base);
    float4 mk = *reinterpret_cast<const float4*>(mask + base);
    float4 pr = *reinterpret_cast<const float4*>(pred + base);
    float4 gv = *reinterpret_cast<const float4*>(gt   + base);

    #pragma unroll
    for (int s = 0; s < ITERS; ++s) {
        // prefetch next iteration (last iter re-reads current; dead loads DCE'd)
        const int sn   = (s + 1 < ITERS) ? (s + 1) : s;
        const int idxn = base + sn * 8;
        float4 nx1 = *reinterpret_cast<const float4*>(in1  + idxn);
        float4 nx2 = *reinterpret_cast<const float4*>(in2  + idxn);
        float4 nmk = *reinterpret_cast<const float4*>(mask + idxn);
        float4 npr = *reinterpret_cast<const float4*>(pred + idxn);
        float4 ngv = *reinterpret_cast<const float4*>(gt   + idxn);
        // keep the 5 prefetch loads issued ahead of this iteration's compute
        __builtin_amdgcn_sched_barrier(0);

        float s1a = sigf(x1.x), s1b = sigf(x1.y), s1c = sigf(x1.z), s1d = sigf(x1.w);
        float s2a = sigf(x2.x), s2b = sigf(x2.y), s2c = sigf(x2.z), s2d = sigf(x2.w);

        // Two slot-permuted K=4 WMMAs cover all 8 k's of this chunk:
        //  step A (x,y): half0 -> k {c,c+1};   half1 -> k {c+4,c+5}
        //  step B (z,w): half0 -> k {c+2,c+3}; half1 -> k {c+6,c+7}
        // A and B operands use identical slot->k mapping, so D[m,n] += f1[m,k]*f2[n,k].
        v2f aA = {s1a, s1b};
        v2f bA = {s2a, s2b};
        acc0 = __builtin_amdgcn_wmma_f32_16x16x4_f32(
            false, aA, false, bA, (short)0, acc0, false, false);
        v2f aB = {s1c, s1d};
        v2f bB = {s2c, s2d};
        acc1 = __builtin_amdgcn_wmma_f32_16x16x4_f32(
            false, aB, false, bB, (short)0, acc1, false, false);

        // elementwise reductions: each loaded element contributes exactly once
        sq1 += s1a * s1a + s1b * s1b + s1c * s1c + s1d * s1d;
        sq2 += s2a * s2a + s2b * s2b + s2c * s2c + s2d * s2d;
        float d0 = mk.x * (s1a - s2a);
        float d1 = mk.y * (s1b - s2b);
        float d2 = mk.z * (s1c - s2c);
        float d3 = mk.w * (s1d - s2d);
        md += d0 * d0 + d1 * d1 + d2 * d2 + d3 * d3;

        float pa = sigf(pr.x), pb = sigf(pr.y), pc = sigf(pr.z), pd = sigf(pr.w);
        sp  += (pa + pb) + (pc + pd);
        sg  += (gv.x + gv.y) + (gv.z + gv.w);
        spg += pa * gv.x + pb * gv.y + pc * gv.z + pd * gv.w;

        x1 = nx1; x2 = nx2; mk = nmk; pr = npr; gv = ngv;
    }

    // --- full-wave reduce of dice sums ---
    #pragma unroll
    for (int off = 16; off >= 1; off >>= 1) {
        spg += __shfl_xor(spg, off, 32);
        sp  += __shfl_xor(sp,  off, 32);
        sg  += __shfl_xor(sg,  off, 32);
    }
    if (lane == 0) {
        atomicAdd(&sDice[0], spg);
        atomicAdd(&sDice[1], sp);
        atomicAdd(&sDice[2], sg);
    }

    // --- per-row sums: lanes L and L+16 share a row ---
    sq1 += __shfl_xor(sq1, 16, 32);
    sq2 += __shfl_xor(sq2, 16, 32);
    md  += __shfl_xor(md,  16, 32);
    if (lane < 16) {
        atomicAdd(&sSq1[row], sq1);
        atomicAdd(&sSq2[row], sq2);
        atomicAdd(&sMD[row],  md);
    }

    // --- Gram C/D layout: vgpr r, lane<16 -> (M=r,N=lane); lane>=16 -> (M=r+8,N=lane-16)
    {
        const int n = lane & 15;
        const int mbase = (lane >> 4) * 8;
        #pragma unroll
        for (int r = 0; r < 8; ++r) {
            atomicAdd(&sGram[(mbase + r) * 16 + n], acc0[r] + acc1[r]);
        }
    }
    __syncthreads();

    // --- block -> global ---
    for (int i = threadIdx.x; i < 256; i += blockDim.x)
        atomicAdd(&ws[i], sGram[i]);
    if (threadIdx.x < 16) {
        atomicAdd(&ws[256 + threadIdx.x], sSq1[threadIdx.x]);
        atomicAdd(&ws[272 + threadIdx.x], sSq2[threadIdx.x]);
        atomicAdd(&ws[288 + threadIdx.x], sMD[threadIdx.x]);
    }
    if (threadIdx.x < 3)
        atomicAdd(&ws[304 + threadIdx.x], sDice[threadIdx.x]);
}

__global__ void finalize_kernel(const float* __restrict__ ws, float* __restrict__ out) {
    __shared__ float sLoss[16];
    const int t = threadIdx.x;
    const float invN = 1.0f / (float)NPER;
    if (t < 16) {
        float simpos = __expf(-(ws[288 + t] * invN) / TAU);
        float sq1 = ws[256 + t] * invN;
        float simneg = 0.0f;
        #pragma unroll
        for (int j = 0; j < 16; ++j) {
            if (j == t) continue;
            float mse = sq1 + ws[272 + j] * invN - 2.0f * ws[t * 16 + j] * invN;
            simneg += __expf(-mse / TAU);
        }
        sLoss[t] = -__logf(simpos * __builtin_amdgcn_rcpf(simpos + simneg));
    }
    __syncthreads();
    if (t == 0) {
        float c = 0.0f;
        #pragma unroll
        for (int i = 0; i < 16; ++i) c += sLoss[i];
        c *= (1.0f / 16.0f);
        float spg = ws[304], sp = ws[305], sg = ws[306];
        float dice = 1.0f - (2.0f * spg + DICE_SMOOTH) *
                            __builtin_amdgcn_rcpf(sp + sg + DICE_SMOOTH);
        out[0] = dice + c;   // total (WEIGHT=1, labeled contrast term = 0)
        out[1] = dice;
        out[2] = 0.0f;       // loss_contrast_labeled
        out[3] = c;          // loss_contrast_unlabeled
    }
}

extern "C" void kernel_launch(void* const* d_in, const int* in_sizes, int n_in,
                              void* d_out, int out_size, void* d_ws, size_t ws_size,
                              hipStream_t stream) {
    const float* pred = (const float*)d_in[0];
    const float* gt   = (const float*)d_in[1];
    const float* in1  = (const float*)d_in[2];
    const float* in2  = (const float*)d_in[3];
    const float* mask = (const float*)d_in[4];
    float* ws  = (float*)d_ws;
    float* out = (float*)d_out;

    zero_ws_kernel<<<1, 512, 0, stream>>>(ws);
    fused_reduce_kernel<<<GRID, BLOCK, 0, stream>>>(pred, gt, in1, in2, mask, ws);
    finalize_kernel<<<1, 32, 0, stream>>>(ws, out);
}